// Attention_61229053772048
// MI455X (gfx1250) — compile-verified
//
#include <hip/hip_runtime.h>
#include <hip/hip_bf16.h>

typedef float v2f __attribute__((ext_vector_type(2)));
typedef float v8f __attribute__((ext_vector_type(8)));

#define NC   1048576   // N*C per batch  (4096*256)
#define NTOK 4096
#define CDIM 256

__device__ __forceinline__ v8f wmma4(v2f a, v2f b, v8f c) {
  // D = A(16x4 f32) * B(4x16 f32) + C(16x16 f32)
  return __builtin_amdgcn_wmma_f32_16x16x4_f32(false, a, false, b, (short)0, c,
                                               false, false);
}

// ---- CDNA5 async global->LDS copy (16B per lane), with safe fallback -------
#if __has_builtin(__builtin_amdgcn_global_load_async_to_lds_b128)
#define HAVE_ASYNC_LDS 1
typedef int v4i_ __attribute__((__vector_size__(4 * sizeof(int))));
typedef __attribute__((address_space(1))) v4i_* g_v4i_ptr;
typedef __attribute__((address_space(3))) v4i_* l_v4i_ptr;
__device__ __forceinline__ void async_cp16(const float* g, float* l) {
  __builtin_amdgcn_global_load_async_to_lds_b128((g_v4i_ptr)g, (l_v4i_ptr)l, 0, 0);
}
__device__ __forceinline__ void async_wait() {
#if __has_builtin(__builtin_amdgcn_s_wait_asynccnt)
  __builtin_amdgcn_s_wait_asynccnt(0);
#else
  asm volatile("s_wait_asynccnt 0x0" ::: "memory");
#endif
}
#else
#define HAVE_ASYNC_LDS 0
__device__ __forceinline__ void async_cp16(const float* g, float* l) {
  *(float4*)l = *(const float4*)g;   // global_load_b128 + ds_store_b128
}
__device__ __forceinline__ void async_wait() {}
#endif

// ---------------------------------------------------------------- K1: QKV GEMM
// Y = x @ W^T,  x:(B*N, 256), W:(256,256).  One wave -> one 16x16 tile.
// All 4 waves of a block share the same 16-row x tile -> stage it in LDS once
// via async-to-LDS. grid (2048, 4, 3), block 128 (4 waves).
__global__ void qkv_gemm_kernel(const float* __restrict__ x,
                                const float* __restrict__ Wq,
                                const float* __restrict__ Wk,
                                const float* __restrict__ Wv,
                                float* __restrict__ Qb,
                                float* __restrict__ Kb,
                                float* __restrict__ Vb) {
  __shared__ __align__(16) float tileX[16 * CDIM];   // 16 KB

  const float* W;
  float* Y;
  if (blockIdx.z == 0)      { W = Wq; Y = Qb; }
  else if (blockIdx.z == 1) { W = Wk; Y = Kb; }
  else                      { W = Wv; Y = Vb; }

  const int t     = threadIdx.x;
  const int wave  = t >> 5;
  const int lane  = t & 31;
  const int row16 = lane & 15;
  const int kb    = (lane >> 4) << 1;      // 0 or 2
  const int m0    = blockIdx.x * 16;
  const int o0    = (blockIdx.y * 4 + wave) * 16;

  // Stage the contiguous 16x256 x tile (4096 floats) into LDS.
  const float* xsrc = x + m0 * CDIM;
#pragma unroll
  for (int r = 0; r < 8; ++r) {
    const int i = r * 128 + t;             // 0..1023, 4 floats each
    async_cp16(xsrc + i * 4, &tileX[i * 4]);
  }
  async_wait();
  __syncthreads();

  const float* arow = &tileX[row16 * CDIM + kb];      // A[m][k] from LDS
  const float* brow = W + (o0 + row16) * CDIM + kb;   // B[k][n] = W[o0+n][k]
  v8f acc = {};
#pragma unroll 8
  for (int k = 0; k < CDIM; k += 4) {
    v2f a = *(const v2f*)(arow + k);
    v2f b = *(const v2f*)(brow + k);
    acc = wmma4(a, b, acc);
  }
  const int mrb = (lane >> 4) * 8;
  float* out = Y + (m0 + mrb) * CDIM + o0 + row16;
#pragma unroll
  for (int r = 0; r < 8; ++r) out[r * CDIM] = acc[r];
}

// ---------------------------------------------------- K2: key softmax (axis=N)
// One block (256 threads) per contiguous 4096-float row. grid 2048.
__global__ void key_softmax_kernel(float* __restrict__ Kb) {
  __shared__ float smax[8];
  __shared__ float ssum[8];
  const int t = threadIdx.x;
  float* p = Kb + blockIdx.x * NTOK;
  float v[16];
#pragma unroll
  for (int i = 0; i < 16; ++i) v[i] = p[t + (i << 8)];
  float m = v[0];
#pragma unroll
  for (int i = 1; i < 16; ++i) m = fmaxf(m, v[i]);
#pragma unroll
  for (int off = 16; off > 0; off >>= 1) m = fmaxf(m, __shfl_xor(m, off, 32));
  if ((t & 31) == 0) smax[t >> 5] = m;
  __syncthreads();
#pragma unroll
  for (int w = 0; w < 8; ++w) m = fmaxf(m, smax[w]);
  float s = 0.f;
#pragma unroll
  for (int i = 0; i < 16; ++i) { v[i] = __expf(v[i] - m); s += v[i]; }
#pragma unroll
  for (int off = 16; off > 0; off >>= 1) s += __shfl_xor(s, off, 32);
  if ((t & 31) == 0) ssum[t >> 5] = s;
  __syncthreads();
  s = 0.f;
#pragma unroll
  for (int w = 0; w < 8; ++w) s += ssum[w];
  const float inv = 1.0f / s;
#pragma unroll
  for (int i = 0; i < 16; ++i) p[t + (i << 8)] = v[i] * inv;
}

// ------------------------------------------------- K3: query softmax (axis=hd)
// One thread per (b,h,n): 32 elements, stride 4096. grid 1024, block 256.
__global__ void query_softmax_kernel(float* __restrict__ Qb) {
  const int idx = blockIdx.x * blockDim.x + threadIdx.x;  // 0..262143
  const int b   = idx >> 15;
  const int rem = idx & 32767;
  const int h   = rem >> 12;
  const int n   = rem & 4095;
  float* p = Qb + b * NC + h * (32 * NTOK) + n;
  float v[32];
#pragma unroll
  for (int d = 0; d < 32; ++d) v[d] = p[d * NTOK];
  float m = v[0];
#pragma unroll
  for (int d = 1; d < 32; ++d) m = fmaxf(m, v[d]);
  float s = 0.f;
#pragma unroll
  for (int d = 0; d < 32; ++d) { v[d] = __expf(v[d] - m); s += v[d]; }
  const float inv = 1.0f / s;
#pragma unroll
  for (int d = 0; d < 32; ++d) p[d * NTOK] = v[d] * inv;
}

// -------------------------------------------- K4: context = key_sm @ V^T (32x32)
// One block per (b,h); 8 waves split K=4096; 4 WMMA accumulators per wave.
__global__ void context_kernel(const float* __restrict__ Ks,
                               const float* __restrict__ Vb,
                               float* __restrict__ ctx) {
  __shared__ float acc_sm[32 * 32];
  const int t = threadIdx.x;
  for (int i = t; i < 1024; i += 256) acc_sm[i] = 0.f;
  __syncthreads();

  const int wave  = t >> 5;
  const int lane  = t & 31;
  const int row16 = lane & 15;
  const int kofs  = (lane >> 4) << 1;
  const int hb    = blockIdx.x * (32 * NTOK);   // (b*8+h)*131072

  const float* k0 = Ks + hb + (row16)      * NTOK + kofs;   // d = 0..15
  const float* k1 = Ks + hb + (16 + row16) * NTOK + kofs;   // d = 16..31
  const float* v0 = Vb + hb + (row16)      * NTOK + kofs;   // e = 0..15
  const float* v1 = Vb + hb + (16 + row16) * NTOK + kofs;   // e = 16..31

  v8f a00 = {}, a01 = {}, a10 = {}, a11 = {};
  const int kbeg = wave * 512, kend = kbeg + 512;
#pragma unroll 4
  for (int k = kbeg; k < kend; k += 4) {
    v2f ka = *(const v2f*)(k0 + k);
    v2f kc = *(const v2f*)(k1 + k);
    v2f va = *(const v2f*)(v0 + k);
    v2f vc = *(const v2f*)(v1 + k);
    a00 = wmma4(ka, va, a00);
    a01 = wmma4(ka, vc, a01);
    a10 = wmma4(kc, va, a10);
    a11 = wmma4(kc, vc, a11);
  }
  const int mrb = (lane >> 4) * 8;
#pragma unroll
  for (int r = 0; r < 8; ++r) {
    atomicAdd(&acc_sm[(mrb + r)      * 32 + row16],      a00[r]);
    atomicAdd(&acc_sm[(mrb + r)      * 32 + 16 + row16], a01[r]);
    atomicAdd(&acc_sm[(16 + mrb + r) * 32 + row16],      a10[r]);
    atomicAdd(&acc_sm[(16 + mrb + r) * 32 + 16 + row16], a11[r]);
  }
  __syncthreads();
  float* out = ctx + blockIdx.x * 1024;
  for (int i = t; i < 1024; i += 256) out[i] = acc_sm[i];
}

// ---------------------- K5: fold Wp into ctx:  M[b][o][h*32+d] = sum_e Wp[o,h*32+e]*ctx[b,h,d,e]
__global__ void fold_kernel(const float* __restrict__ Wp,
                            const float* __restrict__ ctx,
                            float* __restrict__ Mb) {
  const int idx = blockIdx.x * 256 + threadIdx.x;  // b*65536 + o*256 + c
  const int b = idx >> 16;
  const int rem = idx & 65535;
  const int o = rem >> 8;
  const int c = rem & 255;
  const int h = c >> 5, d = c & 31;
  const float* wrow = Wp + o * CDIM + h * 32;
  const float* crow = ctx + ((b * 8 + h) * 32 + d) * 32;
  float s = 0.f;
#pragma unroll
  for (int e = 0; e < 32; ++e) s = fmaf(wrow[e], crow[e], s);
  Mb[idx] = s;
}

// ------------------ K6: out = M[b] @ Qsm[b] + bp + x  (per batch 256x4096, K=256)
// All 4 waves of a block share the same 256x16 Qsm tile -> stage in LDS via
// async-to-LDS (coalesced 64B row segments). grid (256, 4, 8), block 128.
__global__ void proj_gemm_kernel(const float* __restrict__ Mb,
                                 const float* __restrict__ Qs,
                                 const float* __restrict__ bp,
                                 const float* __restrict__ x,
                                 float* __restrict__ out) {
  __shared__ __align__(16) float tileQ[CDIM * 16];   // [k][n], 16 KB

  const int t     = threadIdx.x;
  const int b     = blockIdx.z;
  const int wave  = t >> 5;
  const int lane  = t & 31;
  const int row16 = lane & 15;
  const int kb    = (lane >> 4) << 1;
  const int n0    = blockIdx.x * 16;
  const int o0    = (blockIdx.y * 4 + wave) * 16;

  // Stage Qsm[b][0:256][n0:n0+16] into LDS: 1024 16B segments, 8 rounds.
  const float* qsrc = Qs + b * NC + n0;
#pragma unroll
  for (int r = 0; r < 8; ++r) {
    const int i    = r * 128 + t;          // 0..1023
    const int krow = i >> 2;
    const int seg  = i & 3;
    async_cp16(qsrc + krow * NTOK + seg * 4, &tileQ[i * 4]);
  }
  async_wait();
  __syncthreads();

  const float* arow = Mb + b * 65536 + (o0 + row16) * CDIM + kb;
  v8f acc = {};
#pragma unroll 8
  for (int k = 0; k < CDIM; k += 4) {
    v2f a = *(const v2f*)(arow + k);
    v2f bf;
    bf.x = tileQ[(k + kb)     * 16 + row16];
    bf.y = tileQ[(k + kb + 1) * 16 + row16];
    acc = wmma4(a, bf, acc);
  }
  const int mrb   = (lane >> 4) * 8;
  const int obase = b * NC;
#pragma unroll
  for (int r = 0; r < 8; ++r) {
    const int o   = o0 + mrb + r;
    const int idx = obase + o * NTOK + n0 + row16;
    out[idx] = acc[r] + bp[o] + x[idx];
  }
}

// ---------------------------------------------------------------------- launch
extern "C" void kernel_launch(void* const* d_in, const int* in_sizes, int n_in,
                              void* d_out, int out_size, void* d_ws, size_t ws_size,
                              hipStream_t stream) {
  const float* x  = (const float*)d_in[0];
  const float* Wq = (const float*)d_in[1];
  const float* Wk = (const float*)d_in[2];
  const float* Wv = (const float*)d_in[3];
  const float* Wp = (const float*)d_in[4];
  const float* bp = (const float*)d_in[5];
  float* out = (float*)d_out;

  const int B = 8;
  float* Qb  = (float*)d_ws;        // B*NC
  float* Kb  = Qb  + (size_t)B * NC;
  float* Vb  = Kb  + (size_t)B * NC;
  float* ctx = Vb  + (size_t)B * NC;          // B*8*32*32 = 65536
  float* Mb  = ctx + 65536;                   // B*256*256 = 524288

  // 1) Q/K/V = x @ W^T
  qkv_gemm_kernel<<<dim3(2048, 4, 3), 128, 0, stream>>>(x, Wq, Wk, Wv, Qb, Kb, Vb);
  // 2) softmax over tokens on K (2048 contiguous rows of 4096)
  key_softmax_kernel<<<dim3(2048), 256, 0, stream>>>(Kb);
  // 3) softmax over channels (32, stride 4096) on Q
  query_softmax_kernel<<<dim3(1024), 256, 0, stream>>>(Qb);
  // 4) per-head 32x32 context
  context_kernel<<<dim3(64), 256, 0, stream>>>(Kb, Vb, ctx);
  // 5) fold Wp with ctx -> per-batch 256x256 matrix
  fold_kernel<<<dim3(2048), 256, 0, stream>>>(Wp, ctx, Mb);
  // 6) out = M[b] @ Qsm[b] + bp + residual(x raw-reshape)
  proj_gemm_kernel<<<dim3(256, 4, 8), 128, 0, stream>>>(Mb, Qb, bp, x, out);
}